// detector_26276609917382
// MI455X (gfx1250) — compile-verified
//
#include <hip/hip_runtime.h>

// Detector post-process: per-class bbox decode + NMS.
// B=32, R=300, C=37 -> out (32, 36, 300, 5).
// One workgroup per (batch, class) pair: 32*36 = 1152 blocks, 256 threads.
// CDNA5 paths used: global_load_async_to_lds_{b128,b32} staging (ASYNCcnt /
// s_wait_asynccnt), wave32 shuffle-driven serial NMS recurrence.

#define B_ 32
#define R_ 300
#define C_ 37
#define NC 36               // C-1 output classes
#define NWORDS 10           // ceil(300/32)
#define NTHR 256
#define SCORE_THR_ 0.1f
#define NMS_THR_ 0.4f

__global__ __launch_bounds__(NTHR)
void detector_nms_kernel(const float* __restrict__ rois,
                         const float* __restrict__ pred,
                         const float* __restrict__ scores,
                         const float* __restrict__ im_info,
                         float* __restrict__ out)
{
    __shared__ __align__(16) float rraw[R_][4];   // roi rows (async b128)
    __shared__ __align__(16) float praw[R_][4];   // per-class deltas (async b128)
    __shared__ float osc[R_];                     // per-class scores (async b32)
    __shared__ float ox1[R_], oy1[R_], ox2[R_], oy2[R_];   // decoded, original order
    __shared__ float sx1[R_], sy1[R_], sx2[R_], sy2[R_], ss[R_]; // sorted
    __shared__ int   order[R_];
    __shared__ unsigned supmask[R_][NWORDS];      // suppression bitmatrix (sorted idx)
    __shared__ unsigned keepw[NWORDS];
    __shared__ unsigned char keepO[R_];           // keep flag, original order
    __shared__ int top_r;

    const int tid = threadIdx.x;
    const int bc  = blockIdx.x;
    const int b   = bc / NC;
    const int ci  = bc % NC;       // output class index
    const int c   = ci + 1;        // source class index (class 0 dropped)

    // ---- Stage inputs into LDS via CDNA5 async global->LDS DMA ----
    for (int r = tid; r < R_; r += NTHR) {
        unsigned long long ga_rois = (unsigned long long)(rois  + ((size_t)(b * R_ + r)) * 4);
        unsigned long long ga_pred = (unsigned long long)(pred  + ((size_t)(b * R_ + r)) * (4 * C_) + 4 * c);
        unsigned long long ga_sc   = (unsigned long long)(scores + ((size_t)(b * R_ + r)) * C_ + c);
        unsigned lo_rois = (unsigned)(unsigned long long)&rraw[r][0];
        unsigned lo_pred = (unsigned)(unsigned long long)&praw[r][0];
        unsigned lo_sc   = (unsigned)(unsigned long long)&osc[r];
        asm volatile("global_load_async_to_lds_b128 %0, %1, off"
                     :: "v"(lo_rois), "v"(ga_rois) : "memory");
        asm volatile("global_load_async_to_lds_b128 %0, %1, off"
                     :: "v"(lo_pred), "v"(ga_pred) : "memory");
        asm volatile("global_load_async_to_lds_b32 %0, %1, off"
                     :: "v"(lo_sc), "v"(ga_sc) : "memory");
    }
    asm volatile("s_wait_asynccnt 0x0" ::: "memory");
    __syncthreads();

    const float Hm1   = im_info[b * 3 + 0] - 1.0f;
    const float Wm1   = im_info[b * 3 + 1] - 1.0f;
    const float scale = im_info[2];   // reference uses im_info[0, 2] for all batches

    // ---- Decode + clip + rescale (original roi order) ----
    for (int r = tid; r < R_; r += NTHR) {
        float x1 = rraw[r][0], y1 = rraw[r][1], x2 = rraw[r][2], y2 = rraw[r][3];
        float w  = x2 - x1 + 1.0f;
        float h  = y2 - y1 + 1.0f;
        float cx = x1 + 0.5f * w;
        float cy = y1 + 0.5f * h;
        float dx = praw[r][0] * 0.1f;
        float dy = praw[r][1] * 0.1f;
        float dw = praw[r][2] * 0.2f;
        float dh = praw[r][3] * 0.2f;
        float pcx = dx * w + cx;
        float pcy = dy * h + cy;
        float pw  = expf(dw) * w;
        float ph  = expf(dh) * h;
        float px1 = fminf(fmaxf(pcx - 0.5f * pw, 0.0f), Wm1);
        float py1 = fminf(fmaxf(pcy - 0.5f * ph, 0.0f), Hm1);
        float px2 = fminf(fmaxf(pcx + 0.5f * pw, 0.0f), Wm1);
        float py2 = fminf(fmaxf(pcy + 0.5f * ph, 0.0f), Hm1);
        ox1[r] = px1 / scale; oy1[r] = py1 / scale;
        ox2[r] = px2 / scale; oy2[r] = py2 / scale;
    }
    __syncthreads();

    // ---- Stable descending rank sort (matches jnp.argsort(-s) tie rules) ----
    for (int i = tid; i < R_; i += NTHR) {
        float si = osc[i];
        int rank = 0;
        for (int j = 0; j < R_; ++j) {
            float sj = osc[j];
            rank += (sj > si) || ((sj == si) && (j < i));
        }
        order[rank] = i;
        sx1[rank] = ox1[i]; sy1[rank] = oy1[i];
        sx2[rank] = ox2[i]; sy2[rank] = oy2[i];
        ss[rank]  = si;
    }
    __syncthreads();

    // ---- Initial validity bits + suppression bitmatrix ----
    if (tid < NWORDS) {
        unsigned w = 0u;
        for (int bit = 0; bit < 32; ++bit) {
            int k = tid * 32 + bit;
            if (k < R_ && ss[k] > SCORE_THR_) w |= (1u << bit);
        }
        keepw[tid] = w;
    }
    for (int i = tid; i < R_; i += NTHR) {
        unsigned m[NWORDS];
        #pragma unroll
        for (int w = 0; w < NWORDS; ++w) m[w] = 0u;
        float ax1 = sx1[i], ay1 = sy1[i], ax2 = sx2[i], ay2 = sy2[i];
        float aarea = (ax2 - ax1) * (ay2 - ay1);
        for (int j = i + 1; j < R_; ++j) {
            float bx1 = sx1[j], by1 = sy1[j], bx2 = sx2[j], by2 = sy2[j];
            float barea = (bx2 - bx1) * (by2 - by1);
            float iw = fmaxf(fminf(ax2, bx2) - fmaxf(ax1, bx1), 0.0f);
            float ih = fmaxf(fminf(ay2, by2) - fmaxf(ay1, by1), 0.0f);
            float inter = iw * ih;
            float iou = inter / (aarea + barea - inter + 1e-12f);
            if (iou > NMS_THR_) m[j >> 5] |= (1u << (j & 31));
        }
        #pragma unroll
        for (int w = 0; w < NWORDS; ++w) supmask[i][w] = m[w];
    }
    __syncthreads();

    // ---- Serial suppression recurrence on one wave32 ----
    // Lane w (<10) owns keep word w; liveness of row i broadcast via shuffle.
    if (tid < 32) {
        unsigned kw = (tid < NWORDS) ? keepw[tid] : 0u;
        for (int i = 0; i < R_; ++i) {
            unsigned wi = __shfl(kw, i >> 5, 32);
            if ((wi >> (i & 31)) & 1u) {           // wave-uniform branch
                if (tid < NWORDS) kw &= ~supmask[i][tid];
            }
        }
        if (tid < NWORDS) keepw[tid] = kw;
    }
    __syncthreads();

    // ---- Scatter keep bits back to original roi order ----
    for (int k = tid; k < R_; k += NTHR) {
        unsigned bit = (keepw[k >> 5] >> (k & 31)) & 1u;
        keepO[order[k]] = (unsigned char)bit;
    }
    __syncthreads();

    // ---- Class 0 special case: keep only the top-scoring kept box ----
    if (ci == 0) {
        if (tid == 0) {
            int kfirst = -1;
            for (int w = 0; w < NWORDS && kfirst < 0; ++w) {
                unsigned v = keepw[w];
                if (v) kfirst = w * 32 + (__ffs(v) - 1);
            }
            top_r = (kfirst >= 0) ? order[kfirst] : -1;
        }
        __syncthreads();
        for (int r = tid; r < R_; r += NTHR)
            keepO[r] = (r == top_r) ? (unsigned char)1 : (unsigned char)0;
        __syncthreads();
    }

    // ---- Emit dets = clip(concat(box, score), 0) * keep ----
    float* outp = out + ((size_t)(b * NC + ci)) * (R_ * 5);
    for (int idx = tid; idx < R_ * 5; idx += NTHR) {
        int r = idx / 5;
        int comp = idx - r * 5;
        float v;
        switch (comp) {
            case 0: v = ox1[r]; break;
            case 1: v = oy1[r]; break;
            case 2: v = ox2[r]; break;
            case 3: v = oy2[r]; break;
            default: v = osc[r]; break;
        }
        v = fmaxf(v, 0.0f);
        outp[idx] = keepO[r] ? v : 0.0f;
    }
}

extern "C" void kernel_launch(void* const* d_in, const int* in_sizes, int n_in,
                              void* d_out, int out_size, void* d_ws, size_t ws_size,
                              hipStream_t stream) {
    const float* rois    = (const float*)d_in[0];   // (B, R, 4)
    const float* pred    = (const float*)d_in[1];   // (B, R, 4*C)
    const float* scores  = (const float*)d_in[2];   // (B, R, C)
    const float* im_info = (const float*)d_in[3];   // (B, 3)
    float* out = (float*)d_out;                     // (B, 36, R, 5)
    (void)in_sizes; (void)n_in; (void)out_size; (void)d_ws; (void)ws_size;

    dim3 grid(B_ * NC);
    dim3 block(NTHR);
    detector_nms_kernel<<<grid, block, 0, stream>>>(rois, pred, scores, im_info, out);
}